// ConfCrossEntropyLoss_22067541967686
// MI455X (gfx1250) — compile-verified
//
#include <hip/hip_runtime.h>
#include <stdint.h>

#define T_MAX   512
#define BATCH   32
#define CLASSES 1296
#define L_MAX   64
#define N_ROWS  (T_MAX * BATCH)          // 16384 rows of length V

// TDM streaming tiles: 4 rows * 1296 f32 = 20736 B, double buffered (41.5 KB LDS)
#define TILE_ROWS        4
#define TILE_ELEMS       (TILE_ROWS * CLASSES)     // 5184 (< 65536, fits tile_dim0)
#define TILES_PER_BLOCK  8
#define ROWS_PER_BLOCK   (TILE_ROWS * TILES_PER_BLOCK)  // 32
#define STAT_GRID        (N_ROWS / ROWS_PER_BLOCK)      // 512

typedef __attribute__((ext_vector_type(2))) float        v2f;
typedef __attribute__((ext_vector_type(8))) float        v8f;
typedef __attribute__((ext_vector_type(4))) unsigned int v4u;
typedef __attribute__((ext_vector_type(8))) int          v8i;
typedef __attribute__((ext_vector_type(4))) int          v4i;

__device__ __forceinline__ float wave_max(float v) {
#pragma unroll
  for (int o = 16; o; o >>= 1) v = fmaxf(v, __shfl_xor(v, o, 32));
  return v;
}
__device__ __forceinline__ float wave_sum(float v) {
#pragma unroll
  for (int o = 16; o; o >>= 1) v += __shfl_xor(v, o, 32);
  return v;
}

// ---------------------------------------------------------------------------
// Tensor Data Mover: 1-D contiguous copy of `nelem` f32 from global to LDS.
// D# group0/group1 built per CDNA5 ISA:
//   g0: count=1 | lds_addr | global_addr[56:0] | type=2
//   g1: data_size=4B, tensor_dim0 = tile_dim0 = nelem, dim0 stride = nelem
// Tracked by TENSORcnt; completion ordering via s_wait_tensorcnt.
// This toolchain's builtin takes 6 args (extra int32x8 group before cpol).
// ---------------------------------------------------------------------------
__device__ __forceinline__ void tdm_load_1d(const float* gsrc,
                                            unsigned int lds_byte_addr,
                                            unsigned int nelem) {
  unsigned long long ga = (unsigned long long)(uintptr_t)gsrc;
  v4u g0;
  g0[0] = 1u;                                   // count=1 (valid user D#)
  g0[1] = lds_byte_addr;                        // lds_addr [63:32]
  g0[2] = (unsigned int)ga;                     // global_addr [95:64]
  g0[3] = (unsigned int)((ga >> 32) & 0x01FFFFFFull) | (2u << 30); // addr hi | type=2
  v8i g1;
  g1[0] = (int)(2u << 16);                      // data_size = 2 -> 4 bytes
  g1[1] = (int)((nelem & 0xFFFFu) << 16);       // tensor_dim0 [15:0]  (bits 63:48)
  g1[2] = (int)((nelem >> 16) & 0xFFFFu);       // tensor_dim0 [31:16] (bits 79:64)
  g1[3] = (int)((nelem & 0xFFFFu) << 16);       // tile_dim0 (bits 127:112)
  g1[4] = 0;                                    // tile_dim1 = 0 (unused), tile_dim2 = 0
  g1[5] = (int)nelem;                           // tensor_dim0_stride [31:0]
  g1[6] = 0;
  g1[7] = 0;
  v4i z4 = {0, 0, 0, 0};
  v8i z8 = {0, 0, 0, 0, 0, 0, 0, 0};
  __builtin_amdgcn_tensor_load_to_lds(g0, g1, z4, z4, z8, 0);
}

// ---------------------------------------------------------------------------
// Kernel 1: per-row logsumexp of seq_pred.  1 wave/block, TDM double buffer.
// ---------------------------------------------------------------------------
__global__ void __launch_bounds__(32)
seq_lse_kernel(const float* __restrict__ seq, float* __restrict__ lse) {
  __shared__ __align__(128) float tile[2][TILE_ELEMS];
  const int lane = threadIdx.x;
  const int row0 = blockIdx.x * ROWS_PER_BLOCK;

  tdm_load_1d(seq + (size_t)row0 * CLASSES,
              (unsigned int)(uintptr_t)&tile[0][0], TILE_ELEMS);

  for (int t = 0; t < TILES_PER_BLOCK; ++t) {
    if (t + 1 < TILES_PER_BLOCK) {
      tdm_load_1d(seq + (size_t)(row0 + (t + 1) * TILE_ROWS) * CLASSES,
                  (unsigned int)(uintptr_t)&tile[(t + 1) & 1][0], TILE_ELEMS);
      __builtin_amdgcn_s_wait_tensorcnt(1);   // oldest (current) tile landed
    } else {
      __builtin_amdgcn_s_wait_tensorcnt(0);
    }
    const float* tp = &tile[t & 1][0];
#pragma unroll
    for (int rr = 0; rr < TILE_ROWS; ++rr) {
      const float* rowp = tp + rr * CLASSES;
      float m = -__builtin_inff();
      for (int v = lane; v < CLASSES; v += 32) m = fmaxf(m, rowp[v]);
      m = wave_max(m);
      float s = 0.f;
      for (int v = lane; v < CLASSES; v += 32) s += __expf(rowp[v] - m);
      s = wave_sum(s);
      if (lane == 0) lse[row0 + t * TILE_ROWS + rr] = m + __logf(s);
    }
  }
}

// ---------------------------------------------------------------------------
// Kernel 2: forced alignment DP + backtrack. 1 block (1 wave) per batch item.
// Lane l owns DP rows 2l and 2l+1; choice bits packed via wave32 ballot.
// ---------------------------------------------------------------------------
__global__ void __launch_bounds__(32)
align_kernel(const float* __restrict__ seq, const float* __restrict__ lse,
             const int* __restrict__ label, const int* __restrict__ xlen,
             const int* __restrict__ llen, int* __restrict__ targets) {
  __shared__ unsigned int chlds[T_MAX][2];   // [col][row parity] choice bitmasks
  __shared__ int lab_s[L_MAX];

  const int b = blockIdx.x;
  const int lane = threadIdx.x;
  const int C = xlen[b];
  const int R = llen[b];
  const int i0 = 2 * lane, i1 = 2 * lane + 1;
  const int l0 = label[b * L_MAX + i0];
  const int l1 = label[b * L_MAX + i1];
  lab_s[i0] = l0;
  lab_s[i1] = l1;

  const float NI = -__builtin_inff();

  // column j = 0: dp[0] = pb(0,0), others -inf; choices[0] = 0
  float dp0 = (i0 == 0) ? (seq[(size_t)b * CLASSES + l0] - lse[b]) : NI;
  float dp1 = NI;
  if (lane == 0) { chlds[0][0] = 0u; chlds[0][1] = 0u; }

  for (int j = 1; j < T_MAX; ++j) {
    const int rb = j * BATCH + b;
    // pb gathers for this column (independent of dp -> overlaps with compute)
    const float lsj = lse[rb];
    const float p0 = seq[(size_t)rb * CLASSES + l0];
    const float p1 = seq[(size_t)rb * CLASSES + l1];
    const bool in0 = (i0 < R) && (j >= i0) && (j <= C - R + i0);
    const bool in1 = (i1 < R) && (j >= i1) && (j <= C - R + i1);
    const float pb0 = in0 ? (p0 - lsj) : NI;
    const float pb1 = in1 ? (p1 - lsj) : NI;

    // shifted dp (row i-1) from previous column's state
    float up = __shfl_up(dp1, 1, 32);     // dp[2l-1] lives in lane l-1
    const float sh0 = (lane == 0) ? NI : up;
    const float sh1 = dp0;                // dp[2l] (pre-update)

    const bool ch0 = sh0 > dp0;           // strict >, as reference
    const bool ch1 = sh1 > dp1;
    dp0 = (ch0 ? sh0 : dp0) + pb0;
    dp1 = (ch1 ? sh1 : dp1) + pb1;

    const unsigned int m0 = (unsigned int)__ballot(ch0);
    const unsigned int m1 = (unsigned int)__ballot(ch1);
    if (lane == 0) { chlds[j][0] = m0; chlds[j][1] = m1; }
  }
  __syncthreads();

  if (lane == 0) {                        // serial backtrack, exact mirror of ref
    int row = R - 1;
    for (int j = T_MAX - 1; j >= 0; --j) {
      if (j < C) {
        targets[j * BATCH + b] = lab_s[row];
        const unsigned int bits = chlds[j][row & 1];
        row -= (int)((bits >> (row >> 1)) & 1u);
      } else {
        targets[j * BATCH + b] = 0;
      }
    }
  }
}

// ---------------------------------------------------------------------------
// Kernel 3: closed-form smoothed CE per row of pred. Same TDM streaming shape.
// ce = -(conf*lsm[tgt] + smooth*(Sum_lsm - lsm[tgt])),
// Sum_lsm = Sum(pred) - V*lse_pred,  conf = exp(seq[tgt] - lse_seq).
// ---------------------------------------------------------------------------
__global__ void __launch_bounds__(32)
ce_kernel(const float* __restrict__ pred, const float* __restrict__ seq,
          const float* __restrict__ lse, const int* __restrict__ targets,
          const int* __restrict__ xlen, float* __restrict__ cew) {
  __shared__ __align__(128) float tile[2][TILE_ELEMS];
  const int lane = threadIdx.x;
  const int row0 = blockIdx.x * ROWS_PER_BLOCK;

  tdm_load_1d(pred + (size_t)row0 * CLASSES,
              (unsigned int)(uintptr_t)&tile[0][0], TILE_ELEMS);

  for (int t = 0; t < TILES_PER_BLOCK; ++t) {
    if (t + 1 < TILES_PER_BLOCK) {
      tdm_load_1d(pred + (size_t)(row0 + (t + 1) * TILE_ROWS) * CLASSES,
                  (unsigned int)(uintptr_t)&tile[(t + 1) & 1][0], TILE_ELEMS);
      __builtin_amdgcn_s_wait_tensorcnt(1);
    } else {
      __builtin_amdgcn_s_wait_tensorcnt(0);
    }
    const float* tp = &tile[t & 1][0];
#pragma unroll
    for (int rr = 0; rr < TILE_ROWS; ++rr) {
      const int r = row0 + t * TILE_ROWS + rr;
      const float* rowp = tp + rr * CLASSES;
      float m = -__builtin_inff(), sx = 0.f;
      for (int v = lane; v < CLASSES; v += 32) {
        const float x = rowp[v];
        m = fmaxf(m, x);
        sx += x;
      }
      m = wave_max(m);
      sx = wave_sum(sx);
      float s = 0.f;
      for (int v = lane; v < CLASSES; v += 32) s += __expf(rowp[v] - m);
      s = wave_sum(s);
      if (lane == 0) {
        const int tgt = targets[r];
        const float lse_p = m + __logf(s);
        const float lsm_t = rowp[tgt] - lse_p;
        const float Slsm = sx - (float)CLASSES * lse_p;
        const float conf = __expf(seq[(size_t)r * CLASSES + tgt] - lse[r]);
        const float smooth = (1.f - conf) * (1.f / (float)(CLASSES - 1));
        const int tt = r / BATCH, bb = r % BATCH;
        const float ce = (tt < xlen[bb])
                             ? -(conf * lsm_t + smooth * (Slsm - lsm_t))
                             : 0.f;
        cew[r] = ce;
      }
    }
  }
}

// ---------------------------------------------------------------------------
// Kernel 4: deterministic mean of 16384 ce values, finished with one
// V_WMMA_F32_16X16X4_F32 (B = ones => D[m,n] = rowsum; sum(D) = 16 * total,
// independent of the A VGPR layout).  Single wave => EXEC all-ones for WMMA.
// ---------------------------------------------------------------------------
__global__ void __launch_bounds__(32)
reduce_kernel(const float* __restrict__ cew, float* __restrict__ out) {
  const int lane = threadIdx.x;
  float pa = 0.f, pb = 0.f;
  for (int k = 0; k < N_ROWS; k += 64) {   // fixed-order, deterministic
    pa += cew[k + 2 * lane];
    pb += cew[k + 2 * lane + 1];
  }
  v2f a;      a.x = pa;  a.y = pb;         // 64 partials striped into A(16x4)
  v2f ones;   ones.x = 1.f; ones.y = 1.f;  // B(4x16) = all ones
  v8f c = {0.f, 0.f, 0.f, 0.f, 0.f, 0.f, 0.f, 0.f};
  c = __builtin_amdgcn_wmma_f32_16x16x4_f32(
      /*neg_a=*/false, a, /*neg_b=*/false, ones,
      /*c_mod=*/(short)0, c, /*reuse_a=*/false, /*reuse_b=*/false);
  float sd = c[0] + c[1] + c[2] + c[3] + c[4] + c[5] + c[6] + c[7];
#pragma unroll
  for (int o = 16; o; o >>= 1) sd += __shfl_xor(sd, o, 32);
  if (lane == 0) out[0] = sd * (1.0f / 16.0f) * (1.0f / (float)N_ROWS);
}

// ---------------------------------------------------------------------------
extern "C" void kernel_launch(void* const* d_in, const int* in_sizes, int n_in,
                              void* d_out, int out_size, void* d_ws,
                              size_t ws_size, hipStream_t stream) {
  const float* pred = (const float*)d_in[0];   // [T,B,V] f32
  const float* seq  = (const float*)d_in[1];   // [T,B,V] f32
  const int*  label = (const int*)d_in[2];     // [B,L]
  const int*  xlen  = (const int*)d_in[3];     // [B]
  const int*  llen  = (const int*)d_in[4];     // [B]
  float* out = (float*)d_out;

  char* ws = (char*)d_ws;                      // 192 KB used
  float* lse     = (float*)ws;                        // [N_ROWS]
  int*   targets = (int*)(ws + (size_t)N_ROWS * 4);   // [N_ROWS]
  float* cew     = (float*)(ws + (size_t)2 * N_ROWS * 4); // [N_ROWS]

  seq_lse_kernel<<<STAT_GRID, 32, 0, stream>>>(seq, lse);
  align_kernel<<<BATCH, 32, 0, stream>>>(seq, lse, label, xlen, llen, targets);
  ce_kernel<<<STAT_GRID, 32, 0, stream>>>(pred, seq, lse, targets, xlen, cew);
  reduce_kernel<<<1, 32, 0, stream>>>(cew, out);
}